// TimeWarp_76312978915691
// MI455X (gfx1250) — compile-verified
//
#include <hip/hip_runtime.h>

// TimeWarp (SpecAugment bilinear grid_sample with swapped grid axes).
//   out[b,i,j] = (1-wy[j])*C[b,y0[j],i] + wy[j]*C[b,y1[j],i]
//   C[b,r,i]   = (1-wx[i])*S[b,r,x0[i]] + wx[i]*S[b,r,x1[i]]
// Two-stage: (1) build C (2 MB) in d_ws, (2) stage C[b] (64 KB) into LDS via
// CDNA5 GLOBAL_LOAD_ASYNC_TO_LDS_B128 (+ s_wait_asynccnt, ASYNCcnt path) and
// stream the 134 MB output with coalesced non-temporal b128 stores.
// HBM-bound: ~200 MB @ 23.3 TB/s ~ 9 us.

#define BATCH 32
#define NFREQ 128
#define NTIME 8192

typedef float v4f __attribute__((ext_vector_type(4)));  // native vec for NT store

// ---------------- Stage 2: out from LDS-resident C[b] ----------------------
// (placed first so the disasm snippet shows the async-to-LDS sequence)
__global__ __launch_bounds__(256) void tw_apply(const float* __restrict__ cv,
                                                const int* __restrict__ srcp,
                                                const int* __restrict__ dstp,
                                                float* __restrict__ out) {
  __shared__ float lds[NFREQ * NFREQ];  // 64 KB = C[b]
  const int b = blockIdx.x;
  const unsigned tid = threadIdx.x;

  // ---- async-copy C[b] (16384 f32) global -> LDS, 16 x b128 per thread ----
  {
    const float* gbase = cv + ((size_t)b << 14);
    // low 32 bits of a flat pointer to __shared__ == LDS byte offset
    unsigned ldsAddr = (unsigned)(size_t)(&lds[0]) + tid * 16u;
    unsigned vOff = tid * 16u;  // same offset both sides; k via imm offset
    asm volatile(
        "global_load_async_to_lds_b128 %0, %1, %2 offset:0\n\t"
        "global_load_async_to_lds_b128 %0, %1, %2 offset:4096\n\t"
        "global_load_async_to_lds_b128 %0, %1, %2 offset:8192\n\t"
        "global_load_async_to_lds_b128 %0, %1, %2 offset:12288\n\t"
        "global_load_async_to_lds_b128 %0, %1, %2 offset:16384\n\t"
        "global_load_async_to_lds_b128 %0, %1, %2 offset:20480\n\t"
        "global_load_async_to_lds_b128 %0, %1, %2 offset:24576\n\t"
        "global_load_async_to_lds_b128 %0, %1, %2 offset:28672\n\t"
        "global_load_async_to_lds_b128 %0, %1, %2 offset:32768\n\t"
        "global_load_async_to_lds_b128 %0, %1, %2 offset:36864\n\t"
        "global_load_async_to_lds_b128 %0, %1, %2 offset:40960\n\t"
        "global_load_async_to_lds_b128 %0, %1, %2 offset:45056\n\t"
        "global_load_async_to_lds_b128 %0, %1, %2 offset:49152\n\t"
        "global_load_async_to_lds_b128 %0, %1, %2 offset:53248\n\t"
        "global_load_async_to_lds_b128 %0, %1, %2 offset:57344\n\t"
        "global_load_async_to_lds_b128 %0, %1, %2 offset:61440\n\t"
        "s_wait_asynccnt 0"
        :
        : "v"(ldsAddr), "v"(vOff), "s"(gbase)
        : "memory");
  }
  __syncthreads();

  // ---- per-thread time-warp params for its 4 consecutive j ----
  const int srcPt = srcp[0];
  const int dstPt = dstp[0];
  const float lr = (float)dstPt / (float)srcPt;                  // left_ratio
  const float rr = (float)(NTIME - dstPt) / (float)(NTIME - srcPt);

  const int j0 = (int)blockIdx.y * 512 + (int)(tid & 127u) * 4;
  float wy[4], om[4];
  int a0[4], a1[4];
#pragma unroll
  for (int m = 0; m < 4; ++m) {
    const int j = j0 + m;
    const float tf = (float)j;
    float ind = (j < dstPt) ? (tf / lr)
                            : ((float)srcPt + (tf - (float)dstPt) / rr);
    ind = fminf(fmaxf(ind, 0.0f), (float)(NTIME - 1));
    float yp = ind / (float)(NTIME - 1) * (float)(NFREQ - 1);
    yp = fminf(fmaxf(yp, 0.0f), (float)(NFREQ - 1));
    const float yf = floorf(yp);
    const int y0 = (int)yf;
    a0[m] = y0 << 7;                      // y0*128
    a1[m] = min(y0 + 1, NFREQ - 1) << 7;  // y1*128
    wy[m] = yp - yf;
    om[m] = 1.0f - wy[m];
  }

  // ---- stream 64 output rows; lanes contiguous in j => coalesced b128 ----
  const int iBase = (int)(tid >> 7) * 64;
  float* outb = out + ((size_t)b << 20);
#pragma unroll 4
  for (int iL = 0; iL < 64; ++iL) {
    const int i = iBase + iL;
    v4f o;
    o.x = om[0] * lds[a0[0] + i] + wy[0] * lds[a1[0] + i];
    o.y = om[1] * lds[a0[1] + i] + wy[1] * lds[a1[1] + i];
    o.z = om[2] * lds[a0[2] + i] + wy[2] * lds[a1[2] + i];
    o.w = om[3] * lds[a0[3] + i] + wy[3] * lds[a1[3] + i];
    // output is written once, never reread -> non-temporal store
    __builtin_nontemporal_store(o, (v4f*)(outb + ((size_t)i << 13) + (size_t)j0));
  }
}

// ---------------- Stage 1: C[b,r,i] = lerp over time columns ----------------
__global__ __launch_bounds__(256) void tw_colval(const float* __restrict__ S,
                                                 float* __restrict__ cv) {
  const int b = blockIdx.x;
#pragma unroll
  for (int k = 0; k < 8; ++k) {
    const int e = blockIdx.y * 2048 + k * 256 + (int)threadIdx.x;  // r*128+i
    const int r = e >> 7;
    const int i = e & 127;
    // x_pix = i/(nF-1)*(nT-1), clipped (no-op here), border-safe x1
    float xp = (float)i / 127.0f * 8191.0f;
    xp = fminf(fmaxf(xp, 0.0f), 8191.0f);
    const float xf = floorf(xp);
    const int x0 = (int)xf;
    const int x1 = min(x0 + 1, NTIME - 1);
    const float wx = xp - xf;
    const float* row = S + ((size_t)b << 20) + ((size_t)r << 13);
    // gathered input lines are touched once -> non-temporal loads
    const float s0 = __builtin_nontemporal_load(row + x0);
    const float s1 = __builtin_nontemporal_load(row + x1);
    cv[((size_t)b << 14) + (size_t)e] = (1.0f - wx) * s0 + wx * s1;
  }
}

extern "C" void kernel_launch(void* const* d_in, const int* in_sizes, int n_in,
                              void* d_out, int out_size, void* d_ws, size_t ws_size,
                              hipStream_t stream) {
  (void)in_sizes; (void)n_in; (void)out_size; (void)ws_size;
  const float* S = (const float*)d_in[0];     // [32,128,8192] f32
  const int* srcp = (const int*)d_in[1];      // scalar int
  const int* dstp = (const int*)d_in[2];      // scalar int
  float* out = (float*)d_out;                 // [32,128,8192] f32
  float* cv = (float*)d_ws;                   // needs 32*128*128*4 = 2 MB

  tw_colval<<<dim3(BATCH, 8), 256, 0, stream>>>(S, cv);
  tw_apply<<<dim3(BATCH, 16), 256, 0, stream>>>(cv, srcp, dstp, out);
}